// BM25_56358560858178
// MI455X (gfx1250) — compile-verified
//
#include <hip/hip_runtime.h>

// BM25 for MI455X (gfx1250, wave32).
// - Stage query (16KB) + passage (32KB) into LDS with CDNA5 async
//   global->LDS b128 loads (ASYNCcnt-tracked), fenced by s_wait_asynccnt.
// - One thread per query position counts qtf/ptf against LDS (ds_load_b128),
//   computes its BM25 contribution (divided by qtf to dedup).
// - Wave reduction via a load-bearing V_WMMA_F32_16X16X4_F32
//   (A = contrib in both K slots, B = 0.5 => D[m][n] = contrib[m]+contrib[m+16]).
// - Deterministic: no float atomics; fixed-order wave->block->global sums.

typedef float v2f __attribute__((ext_vector_type(2)));
typedef float v8f __attribute__((ext_vector_type(8)));

#define TPB 256
#define MAX_Q 4096
#define MAX_P 8192

// Async global->LDS stage of n ints (16B-vector chunks + scalar tail).
// lds_base = wave-relative LDS byte address (low 32 bits of generic pointer).
__device__ __forceinline__ void async_stage(int* __restrict__ s_dst,
                                            const int* __restrict__ g_src,
                                            int n, int tid) {
  unsigned lds_base = (unsigned)(size_t)&s_dst[0];
  const int n4 = n >> 2;  // int4 chunks
  for (int i = tid; i < n4; i += TPB) {
    unsigned lds_addr = lds_base + ((unsigned)i << 4);
    unsigned goff = (unsigned)i << 4;
    // GVS mode: mem = SGPR64 base + VGPR32 offset; VDST = LDS byte address.
    asm volatile("global_load_async_to_lds_b128 %0, %1, %2"
                 :
                 : "v"(lds_addr), "v"(goff), "s"(g_src)
                 : "memory");
  }
  for (int i = (n4 << 2) + tid; i < n; i += TPB) s_dst[i] = g_src[i];
}

__global__ __launch_bounds__(TPB) void bm25_main_kernel(
    const int* __restrict__ q_ids, const int* __restrict__ p_ids,
    const float* __restrict__ dfs, float* __restrict__ block_sums,
    int n_q, int n_p) {
  __shared__ __align__(16) int s_q[MAX_Q];
  __shared__ __align__(16) int s_p[MAX_P];
  __shared__ float s_wave[TPB / 32];

  const int tid = threadIdx.x;

  // CDNA5 async global->LDS staging (writes LDS directly, no VGPR round-trip).
  async_stage(s_q, q_ids, n_q, tid);
  async_stage(s_p, p_ids, n_p, tid);
  asm volatile("s_wait_asynccnt 0" ::: "memory");  // our wave's async writes done
  __syncthreads();                                 // all waves' writes visible

  const int gid = blockIdx.x * TPB + tid;
  const bool active = gid < n_q;
  const int idx = active ? gid : 0;       // branchless; keeps EXEC full for WMMA
  const int t = s_q[idx];

  // Early gather of document frequency; emit a CDNA5 global prefetch for it.
  __builtin_prefetch(&dfs[t], 0, 1);      // -> global_prefetch_b8
  const float df = dfs[t];

  // Count occurrences of t in query (qtf) and passage (ptf) from LDS, b128 at a time.
  int qtf = 0, ptf = 0;
  {
    const int4* q4 = (const int4*)s_q;
    const int4* p4 = (const int4*)s_p;
    const int nq4 = n_q >> 2;
    const int np4 = n_p >> 2;
    for (int i = 0; i < nq4; ++i) {
      int4 v = q4[i];
      qtf += (v.x == t) + (v.y == t) + (v.z == t) + (v.w == t);
    }
    for (int i = nq4 << 2; i < n_q; ++i) qtf += (s_q[i] == t);
    for (int i = 0; i < np4; ++i) {
      int4 v = p4[i];
      ptf += (v.x == t) + (v.y == t) + (v.z == t) + (v.w == t);
    }
    for (int i = np4 << 2; i < n_p; ++i) ptf += (s_p[i] == t);
  }

  // BM25 terms (K1=1.2, K3=8, B=0.75, N=8841823, L_AVE=55, l_d = n_p).
  const float fq = (float)qtf;            // >= 1 (self-match) -> safe divide
  const float fp = (float)ptf;
  const float term1 = fq / (8.0f + fq);
  const float denom_c = 1.2f * (1.0f - 0.75f + 0.75f * ((float)n_p) / 55.0f);
  const float term2 = 1.2f * fp / (fp + denom_c);
  const float term3 = log2f((8841823.0f - df + 0.5f) / (df + 0.5f));
  // Divide by qtf: token t appears qtf times across query positions, so the
  // per-position sum equals the reference's sum over unique tokens.
  float contrib = active ? (term1 * term2 * term3 / fq) : 0.0f;

  // ---- Wave reduction via V_WMMA_F32_16X16X4_F32 ----
  // A (16x4): lanes 0-15 supply K=0,1 for M=lane; lanes 16-31 supply K=2,3 for
  // M=lane-16. Replicating contrib into both K slots and using B == 0.5
  // everywhere gives D[m][n] = contrib[m] + contrib[m+16] exactly.
  v2f a; a.x = contrib; a.y = contrib;
  v2f b; b.x = 0.5f;    b.y = 0.5f;
  v8f c = {};
  v8f d = __builtin_amdgcn_wmma_f32_16x16x4_f32(
      /*neg_a=*/false, a, /*neg_b=*/false, b,
      /*c_mod=*/(short)0, c, /*reuse_a=*/false, /*reuse_b=*/false);

  // D layout: lanes 0-15 hold rows M=0..7 (one per VGPR), lanes 16-31 rows 8..15.
  float my8 = d[0] + d[1] + d[2] + d[3] + d[4] + d[5] + d[6] + d[7];
  // Combine the two half-wave partial sums -> every lane holds the wave total.
  float wave_total = my8 + __shfl_xor(my8, 16, 32);

  const int lane = tid & 31;
  const int wid = tid >> 5;
  if (lane == 0) s_wave[wid] = wave_total;
  __syncthreads();

  if (tid == 0) {
    float s = 0.0f;
    for (int w = 0; w < TPB / 32; ++w) s += s_wave[w];   // fixed order
    block_sums[blockIdx.x] = s;
  }
}

__global__ void bm25_final_kernel(const float* __restrict__ block_sums,
                                  float* __restrict__ out, int n_blocks) {
  if (blockIdx.x == 0 && threadIdx.x == 0) {
    float s = 0.0f;
    for (int i = 0; i < n_blocks; ++i) s += block_sums[i];  // fixed order
    out[0] = s;
  }
}

extern "C" void kernel_launch(void* const* d_in, const int* in_sizes, int n_in,
                              void* d_out, int out_size, void* d_ws, size_t ws_size,
                              hipStream_t stream) {
  const int*   q_ids = (const int*)d_in[0];    // query_ids  [4096] int32
  const int*   p_ids = (const int*)d_in[1];    // passage_ids[8192] int32
  const float* dfs   = (const float*)d_in[2];  // dfs [8388608] float32

  const int n_q = in_sizes[0];
  const int n_p = in_sizes[1];

  float* block_sums = (float*)d_ws;
  const int n_blocks = (n_q + TPB - 1) / TPB;  // 16 for the reference shapes

  bm25_main_kernel<<<n_blocks, TPB, 0, stream>>>(q_ids, p_ids, dfs,
                                                 block_sums, n_q, n_p);
  bm25_final_kernel<<<1, 32, 0, stream>>>(block_sums, (float*)d_out, n_blocks);
}